// GIN_19181323944513
// MI455X (gfx1250) — compile-verified
//
#include <hip/hip_runtime.h>

#define Nn 100000
#define Ee 1600000
#define HIDD 128
#define Pp 100000
#define Ll 3
#define LDST 132   // padded LDS row stride (floats): 16B-aligned rows, 4-bank row skew

typedef __attribute__((ext_vector_type(2))) float v2f;
typedef __attribute__((ext_vector_type(8))) float v8f;
typedef unsigned int u32x4 __attribute__((ext_vector_type(4)));
typedef int i32x8 __attribute__((ext_vector_type(8)));
typedef int i32x4 __attribute__((ext_vector_type(4)));

// ---------------- TDM: DMA a 128x128 f32 weight tile global->LDS ----------------
// D# per CDNA5 ISA ch.8: group0 = {flags, lds_addr, global_addr, type=2},
// group1 = {data_size=4B + LDS pad (128 DW data + 4 DW pad = 132-float rows),
//           tensor_dim0/1 = 128, tile_dim0/1 = 128, dim0_stride = 128}.
// amdgpu-toolchain (clang-23) form: 6 args (g0, g1, g2, g3, extra i32x8, cpol).
__device__ __forceinline__ void tdm_load_weight_128x128(const float* gsrc, float* lds_dst) {
    unsigned long long ga = (unsigned long long)(size_t)gsrc;
    unsigned int ldsa = (unsigned int)(size_t)lds_dst;   // low 32 bits of generic = LDS offset
    u32x4 g0;
    g0.x = 1u;                                            // count=1 (valid user descriptor)
    g0.y = ldsa;                                          // lds_addr
    g0.z = (unsigned int)(ga & 0xFFFFFFFFull);            // global_addr[31:0]
    g0.w = (unsigned int)((ga >> 32) & 0x1FFFFFFull)      // global_addr[56:32]
         | (2u << 30);                                    // type = 2 ("image")
    i32x8 g1;
    g1[0] = (int)((2u << 16)        // data_size = 4 bytes
                | (1u << 20)        // pad_enable
                | (6u << 22)        // pad_interval: 2^(6+1) = 128 DWORDs
                | (3u << 25));      // pad_amount: 3+1 = 4 DWORDs  -> 132-float rows
    g1[1] = (int)(128u << 16);      // tensor_dim0 = 128 (bits 63:48)
    g1[2] = (int)(128u << 16);      // tensor_dim1 = 128 (bits 95:80)
    g1[3] = (int)(128u << 16);      // tile_dim0   = 128 (bits 127:112)
    g1[4] = 128;                    // tile_dim1   = 128 ; tile_dim2 = 0 (2D)
    g1[5] = 128;                    // tensor_dim0_stride = 128 elements
    g1[6] = 0;
    g1[7] = 0;
    i32x4 z4 = {0, 0, 0, 0};
    i32x8 z8 = {0, 0, 0, 0, 0, 0, 0, 0};
    __builtin_amdgcn_tensor_load_to_lds(g0, g1, z4, z4, z8, 0);
}

// ---------------- utility kernels ----------------

__global__ void zero_kernel(float* __restrict__ p, long n) {
    long i = (long)blockIdx.x * blockDim.x + threadIdx.x;
    long stride = (long)gridDim.x * blockDim.x;
    for (; i < n; i += stride) p[i] = 0.0f;
}

__global__ void deg_kernel(const int* __restrict__ dst, float* __restrict__ deg) {
    int e = blockIdx.x * blockDim.x + threadIdx.x;
    if (e < Ee) atomicAdd(&deg[dst[e]], 1.0f);
}

__global__ void invdeg_kernel(const float* __restrict__ deg, float* __restrict__ invdeg) {
    int n = blockIdx.x * blockDim.x + threadIdx.x;
    if (n < Nn) invdeg[n] = 1.0f / fmaxf(deg[n], 1.0f);
}

// scatter-add mean-aggregation numerator: agg[dst] += h[src]  (float4 per thread)
__global__ void agg_kernel(const float* __restrict__ h, const int* __restrict__ src,
                           const int* __restrict__ dst, float* __restrict__ agg) {
    long idx = (long)blockIdx.x * blockDim.x + threadIdx.x;
    if (idx >= (long)Ee * 32) return;
    int e = (int)(idx >> 5);
    int j = (int)(idx & 31) * 4;
    const float4 v = *(const float4*)(h + (long)src[e] * HIDD + j);
    float* a = agg + (long)dst[e] * HIDD + j;
    atomicAdd(a + 0, v.x);
    atomicAdd(a + 1, v.y);
    atomicAdd(a + 2, v.z);
    atomicAdd(a + 3, v.w);
}

// mu/var -> fused affine: bn(x) = scale*x + shift
__global__ void finalize_kernel(const float* __restrict__ sum, const float* __restrict__ ssq,
                                const float* __restrict__ g, const float* __restrict__ b,
                                float* __restrict__ scale, float* __restrict__ shift,
                                float nrows) {
    int j = threadIdx.x;
    if (j < HIDD) {
        float mu  = sum[j] / nrows;
        float var = fmaxf(ssq[j] / nrows - mu * mu, 0.0f);
        float rs  = rsqrtf(var + 1e-5f);
        float sc  = g[j] * rs;
        scale[j] = sc;
        shift[j] = b[j] - sc * mu;
    }
}

// out = relu(scale*in + shift); optionally accumulate column stats of OUT
__global__ void apply_bn_kernel(const float* __restrict__ in, float* __restrict__ out,
                                const float* __restrict__ scale, const float* __restrict__ shift,
                                float* __restrict__ gSum, float* __restrict__ gSsq,
                                int do_stats, long ntot) {
    __shared__ float sSum[HIDD], sSsq[HIDD];
    int tid = threadIdx.x;
    if (do_stats) {
        if (tid < HIDD) { sSum[tid] = 0.0f; sSsq[tid] = 0.0f; }
        __syncthreads();
    }
    long idx = (long)blockIdx.x * blockDim.x + tid;
    int col = (int)(idx & 127);
    float v = 0.0f;
    if (idx < ntot) {
        v = fmaxf(scale[col] * in[idx] + shift[col], 0.0f);
        out[idx] = v;
    }
    if (do_stats) {
        atomicAdd(&sSum[col], v);
        atomicAdd(&sSsq[col], v * v);
        __syncthreads();
        if (tid < HIDD) {
            atomicAdd(&gSum[tid], sSum[tid]);
            atomicAdd(&gSsq[tid], sSsq[tid]);
        }
    }
}

// ---------------- fused GEMM (+BN stats), TDM-staged weights ----------------
// Y[128-block x 128] = A @ W^T ; A built on the fly:
//   mode 0: A = h + agg * invdeg            (GIN z)
//   mode 1: A = relu(scale * Ain + shift)   (inner BN+ReLU of MLP)
__global__ __launch_bounds__(256)
void gemm_bn_kernel(const float* __restrict__ Ain,
                    const float* __restrict__ Aaux,     // agg   (mode 0)
                    const float* __restrict__ invdeg,   //       (mode 0)
                    const float* __restrict__ scale,    //       (mode 1)
                    const float* __restrict__ shift,    //       (mode 1)
                    const float* __restrict__ W,        // [128][128], row j = out feature
                    float* __restrict__ Yout,
                    float* __restrict__ gSum, float* __restrict__ gSsq,
                    int nrows, int mode) {
    __shared__ __align__(16) float zs[128 * LDST];
    __shared__ __align__(16) float wl[128 * LDST];
    __shared__ float sSum[HIDD];
    __shared__ float sSsq[HIDD];

    int tid = threadIdx.x;
    int rowbase = blockIdx.x * 128;

    // kick off the weight-tile DMA (one wave issues; TDM pads rows to 132 floats)
    if (tid < 32) tdm_load_weight_128x128(W, wl);

    if (tid < HIDD) { sSum[tid] = 0.0f; sSsq[tid] = 0.0f; }

    // concurrently build the activation strip in LDS
    #pragma unroll 4
    for (int t = 0; t < 64; ++t) {
        int i = tid + t * 256;
        int r = i >> 7, c = i & 127;
        int gr = rowbase + r;
        float v = 0.0f;
        if (gr < nrows) {
            float a = Ain[(long)gr * HIDD + c];
            if (mode == 0) v = a + Aaux[(long)gr * HIDD + c] * invdeg[gr];
            else           v = fmaxf(scale[c] * a + shift[c], 0.0f);
        }
        zs[r * LDST + c] = v;
    }
    if (tid < 32) __builtin_amdgcn_s_wait_tensorcnt(0);
    __syncthreads();

    // wave w owns rows [w*16, w*16+16) of the block; 8 column tiles
    int lane = tid & 31;
    int wv   = tid >> 5;
    int lm = lane & 15, hi = lane >> 4;

    v8f acc[8];
    #pragma unroll
    for (int ct = 0; ct < 8; ++ct) acc[ct] = (v8f){0,0,0,0,0,0,0,0};

    const float* za = &zs[(wv * 16 + lm) * LDST + 2 * hi];
    for (int ks = 0; ks < 32; ++ks) {            // K=128 in steps of 4
        v2f a = *(const v2f*)(za + 4 * ks);      // ds_load_b64
        #pragma unroll
        for (int ct = 0; ct < 8; ++ct) {
            v2f b = *(const v2f*)&wl[(ct * 16 + lm) * LDST + 4 * ks + 2 * hi];
            acc[ct] = __builtin_amdgcn_wmma_f32_16x16x4_f32(
                false, a, false, b, (short)0, acc[ct], false, false);
        }
    }

    // epilogue: store Y, accumulate column stats (zeroed rows contribute 0)
    #pragma unroll
    for (int ct = 0; ct < 8; ++ct) {
        int col = ct * 16 + lm;
        float s = 0.0f, ss = 0.0f;
        #pragma unroll
        for (int v = 0; v < 8; ++v) {
            float y = acc[ct][v];
            int gr = rowbase + wv * 16 + v + 8 * hi;   // C/D layout: M = v + 8*hi
            if (gr < nrows) Yout[(long)gr * HIDD + col] = y;
            s += y; ss += y * y;
        }
        atomicAdd(&sSum[col], s);
        atomicAdd(&sSsq[col], ss);
    }
    __syncthreads();
    if (tid < HIDD) {
        atomicAdd(&gSum[tid], sSum[tid]);
        atomicAdd(&gSsq[tid], sSsq[tid]);
    }
}

// ---------------- link predictor ----------------

__device__ __forceinline__ void block_gemm_relu_lds(const float* __restrict__ As,
                                                    const float* __restrict__ Ws,
                                                    float* __restrict__ Out,
                                                    const float* __restrict__ bias,
                                                    int tid) {
    int lane = tid & 31;
    int wv   = tid >> 5;
    int lm = lane & 15, hi = lane >> 4;
    v8f acc[8];
    #pragma unroll
    for (int ct = 0; ct < 8; ++ct) acc[ct] = (v8f){0,0,0,0,0,0,0,0};
    const float* za = &As[(wv * 16 + lm) * LDST + 2 * hi];
    for (int ks = 0; ks < 32; ++ks) {
        v2f a = *(const v2f*)(za + 4 * ks);
        #pragma unroll
        for (int ct = 0; ct < 8; ++ct) {
            v2f b = *(const v2f*)&Ws[(ct * 16 + lm) * LDST + 4 * ks + 2 * hi];
            acc[ct] = __builtin_amdgcn_wmma_f32_16x16x4_f32(
                false, a, false, b, (short)0, acc[ct], false, false);
        }
    }
    #pragma unroll
    for (int ct = 0; ct < 8; ++ct) {
        int col = ct * 16 + lm;
        float bb = bias[col];
        #pragma unroll
        for (int v = 0; v < 8; ++v) {
            Out[(wv * 16 + v + 8 * hi) * LDST + col] = fmaxf(acc[ct][v] + bb, 0.0f);
        }
    }
}

__global__ __launch_bounds__(256)
void predict_kernel(const float* __restrict__ h,
                    const int* __restrict__ si, const int* __restrict__ di,
                    const float* __restrict__ pW1, const float* __restrict__ pb1,
                    const float* __restrict__ pW2, const float* __restrict__ pb2,
                    const float* __restrict__ pW3, const float* __restrict__ pb3,
                    float* __restrict__ out, int npairs) {
    __shared__ __align__(16) float Tld[128 * LDST];
    __shared__ __align__(16) float Uld[128 * LDST];
    __shared__ __align__(16) float Wld[128 * LDST];
    __shared__ float w3s[HIDD];

    int tid = threadIdx.x;
    int pbase = blockIdx.x * 128;

    if (tid < 32) tdm_load_weight_128x128(pW1, Wld);   // DMA W1 while we gather
    if (tid < HIDD) w3s[tid] = pW3[tid];

    // gather t = h[s] * h[d]
    #pragma unroll 4
    for (int t = 0; t < 64; ++t) {
        int i = tid + t * 256;
        int p = i >> 7, k = i & 127;
        int gp = pbase + p;
        int s = 0, d = 0;
        if (gp < npairs) { s = si[gp]; d = di[gp]; }
        Tld[p * LDST + k] = h[(long)s * HIDD + k] * h[(long)d * HIDD + k];
    }
    if (tid < 32) __builtin_amdgcn_s_wait_tensorcnt(0);
    __syncthreads();

    block_gemm_relu_lds(Tld, Wld, Uld, pb1, tid);   // U = relu(T @ W1^T + b1)
    __syncthreads();

    if (tid < 32) {
        tdm_load_weight_128x128(pW2, Wld);          // DMA W2 over W1
        __builtin_amdgcn_s_wait_tensorcnt(0);
    }
    __syncthreads();

    block_gemm_relu_lds(Uld, Wld, Tld, pb2, tid);   // T = relu(U @ W2^T + b2)
    __syncthreads();

    if (tid < HIDD) {
        int gp = pbase + tid;
        float s = 0.0f;
        #pragma unroll 8
        for (int k = 0; k < HIDD; ++k) s += Tld[tid * LDST + k] * w3s[k];
        if (gp < npairs) out[gp] = s + pb3[0];
    }
}

// ---------------- host orchestration ----------------

extern "C" void kernel_launch(void* const* d_in, const int* in_sizes, int n_in,
                              void* d_out, int out_size, void* d_ws, size_t ws_size,
                              hipStream_t stream) {
    (void)in_sizes; (void)n_in; (void)out_size; (void)ws_size;

    const float* x       = (const float*)d_in[0];
    const int*   src     = (const int*)d_in[1];
    const int*   dst     = (const int*)d_in[2];
    const int*   pos_src = (const int*)d_in[3];
    const int*   pos_dst = (const int*)d_in[4];
    const int*   neg_src = (const int*)d_in[5];
    const int*   neg_dst = (const int*)d_in[6];
    const float* W0      = (const float*)d_in[7];
    const float* W1      = (const float*)d_in[8];
    const float* g_mlp   = (const float*)d_in[9];
    const float* b_mlp   = (const float*)d_in[10];
    const float* g_apply = (const float*)d_in[11];
    const float* b_apply = (const float*)d_in[12];
    const float* g_out   = (const float*)d_in[13];
    const float* b_out   = (const float*)d_in[14];
    const float* pW1     = (const float*)d_in[15];
    const float* pb1     = (const float*)d_in[16];
    const float* pW2     = (const float*)d_in[17];
    const float* pb2     = (const float*)d_in[18];
    const float* pW3     = (const float*)d_in[19];
    const float* pb3     = (const float*)d_in[20];
    float* out = (float*)d_out;

    float* w = (float*)d_ws;
    float* deg    = w; w += Nn;
    float* invdeg = w; w += Nn;
    float* hbuf   = w; w += (long)Nn * HIDD;
    float* bufA   = w; w += (long)Nn * HIDD;   // agg, then y1
    float* bufB   = w; w += (long)Nn * HIDD;   // y0
    float* st     = w;                          // 12 * 128 stats floats
    float *sum0 = st,        *ssq0 = st + 128,  *scale0 = st + 256,  *shift0 = st + 384;
    float *sum1 = st + 512,  *ssq1 = st + 640,  *scale1 = st + 768,  *shift1 = st + 896;
    float *sum2 = st + 1024, *ssq2 = st + 1152, *scale2 = st + 1280, *shift2 = st + 1408;

    dim3 b256(256);
    long ntot = (long)Nn * HIDD;
    int  gemm_blocks = (Nn + 127) / 128;

    zero_kernel<<<1024, b256, 0, stream>>>(deg, (long)Nn);
    deg_kernel<<<(Ee + 255) / 256, b256, 0, stream>>>(dst, deg);
    invdeg_kernel<<<(Nn + 255) / 256, b256, 0, stream>>>(deg, invdeg);

    for (int l = 0; l < Ll; ++l) {
        const float* h_in = (l == 0) ? x : hbuf;
        zero_kernel<<<4096, b256, 0, stream>>>(bufA, ntot);
        zero_kernel<<<1, b256, 0, stream>>>(st, 12 * 128);

        long aggwork = (long)Ee * 32;
        agg_kernel<<<(int)((aggwork + 255) / 256), b256, 0, stream>>>(h_in, src, dst, bufA);

        // y0 = (h + agg/deg) @ W0^T  (+ stats for inner BN)
        gemm_bn_kernel<<<gemm_blocks, b256, 0, stream>>>(
            h_in, bufA, invdeg, nullptr, nullptr,
            W0 + (long)l * HIDD * HIDD, bufB, sum0, ssq0, Nn, 0);
        finalize_kernel<<<1, 128, 0, stream>>>(sum0, ssq0, g_mlp + l * HIDD, b_mlp + l * HIDD,
                                               scale0, shift0, (float)Nn);

        // y1 = relu(bn0(y0)) @ W1^T  (+ stats for apply BN)
        gemm_bn_kernel<<<gemm_blocks, b256, 0, stream>>>(
            bufB, nullptr, nullptr, scale0, shift0,
            W1 + (long)l * HIDD * HIDD, bufA, sum1, ssq1, Nn, 1);
        finalize_kernel<<<1, 128, 0, stream>>>(sum1, ssq1, g_apply + l * HIDD, b_apply + l * HIDD,
                                               scale1, shift1, (float)Nn);

        // h_mid = relu(bn1(y1)); collect stats for outer BN if needed
        apply_bn_kernel<<<(int)((ntot + 255) / 256), b256, 0, stream>>>(
            bufA, hbuf, scale1, shift1, sum2, ssq2, (l < Ll - 1) ? 1 : 0, ntot);

        if (l < Ll - 1) {
            finalize_kernel<<<1, 128, 0, stream>>>(sum2, ssq2, g_out + l * HIDD, b_out + l * HIDD,
                                                   scale2, shift2, (float)Nn);
            apply_bn_kernel<<<(int)((ntot + 255) / 256), b256, 0, stream>>>(
                hbuf, hbuf, scale2, shift2, nullptr, nullptr, 0, ntot);
        }
    }

    int pblocks = (Pp + 127) / 128;
    predict_kernel<<<pblocks, b256, 0, stream>>>(hbuf, pos_src, pos_dst,
                                                 pW1, pb1, pW2, pb2, pW3, pb3, out, Pp);
    predict_kernel<<<pblocks, b256, 0, stream>>>(hbuf, neg_src, neg_dst,
                                                 pW1, pb1, pW2, pb2, pW3, pb3, out + Pp, Pp);
}